// CrfDecodeBackwardRnnCell_38809324486882
// MI455X (gfx1250) — compile-verified
//
#include <hip/hip_runtime.h>
#include <stdint.h>

// ---------------------------------------------------------------------------
// CRF backward decode, CDNA5 (gfx1250).
// Strategy: convert a latency-bound HBM pointer chase into a bandwidth-bound
// stream: async-copy backpointer rows into LDS (ASYNCcnt pipeline), chase in
// LDS (16 parallel chains/block, bank-conflict-free padded rows), then gather
// tags with independent non-temporal loads.
// ---------------------------------------------------------------------------

#if __has_builtin(__builtin_amdgcn_global_load_async_to_lds_b128)
#define HAS_ASYNC 1
#else
#define HAS_ASYNC 0
#endif

#define GAS __attribute__((address_space(1)))
#define LAS __attribute__((address_space(3)))

typedef int v4i_t __attribute__((ext_vector_type(4)));

__device__ __forceinline__ void async_cp16(const int* gsrc, int* ldst) {
#if HAS_ASYNC
  // Builtin signature (from clang diagnostic): param 1 is
  // 'int __vector_size__(4) __device__ *'  ==  v4i addrspace(1)* (global src);
  // param 2 is the LDS destination, then imm offset and imm cpol.
  __builtin_amdgcn_global_load_async_to_lds_b128(
      (GAS v4i_t*)gsrc, (LAS v4i_t*)ldst, 0, 0);
#else
  *(int4*)ldst = *(const int4*)gsrc;   // global_load_b128 + ds_store_b128
#endif
}

template <int N>
__device__ __forceinline__ void wait_async_le() {
#if HAS_ASYNC
#if __has_builtin(__builtin_amdgcn_s_wait_asynccnt)
  __builtin_amdgcn_s_wait_asynccnt(N);
#else
  asm volatile("s_wait_asynccnt %0" ::"i"(N) : "memory");
#endif
#endif
}

namespace crf {
constexpr int T  = 256;   // time steps
constexpr int B  = 4096;  // batch (independent chains)
constexpr int NT = 128;   // tags per row (512 B rows)

constexpr int NB   = 16;          // chains per block
constexpr int TC   = 8;           // time steps per staged chunk
constexpr int ROWS = NB * TC;     // 128 rows / chunk
constexpr int CH   = T / TC;      // 32 chunks
constexpr int RS   = NT + 4;      // 132-int padded row: chase lanes hit banks 4*l+s
}  // namespace crf

__global__ __launch_bounds__(256, 2)
void crf_decode_kernel(const float* __restrict__ tags,
                       const int* __restrict__ bp,
                       const int* __restrict__ init_state,
                       float* __restrict__ out) {
  using namespace crf;

  __shared__ int buf[2][ROWS * RS];  // 2 x 67,584 B double buffer
  __shared__ int states[TC][NB];     // pre-update state per (t_local, chain)

  const int tid = threadIdx.x;       // 256 threads = 8 wave32
  const int b0  = blockIdx.x * NB;

  // Staging role: thread -> (row, half-row). Row = 512 B; half = 256 B = 16x b128.
  const int s_row  = tid >> 1;       // 0..127
  const int s_half = tid & 1;        // 0..1
  const int s_tl   = s_row / NB;
  const int s_bl   = s_row % NB;

  int s = 0;
  if (tid < NB) s = init_state[b0 + tid];

  auto stage = [&](int c, int slot) {
    const int t = c * TC + s_tl;
    const int* gsrc = bp + ((size_t)t * B + (size_t)(b0 + s_bl)) * NT
                         + s_half * (NT / 2);
    int* ldst = &buf[slot][s_row * RS + s_half * (NT / 2)];
#pragma unroll
    for (int k = 0; k < NT / 8; ++k)          // 16 x 16B per thread
      async_cp16(gsrc + 4 * k, ldst + 4 * k); // -> 16 async ops per wave per chunk
  };

  stage(0, 0);

  for (int c = 0; c < CH; ++c) {
    const int slot = c & 1;
    if (c + 1 < CH) {
      stage(c + 1, slot ^ 1);   // prefetch next chunk into other buffer
      wait_async_le<16>();      // in-order completion => chunk c has landed
    } else {
      wait_async_le<0>();
    }
    __syncthreads();

    // Chase: 16 chains in parallel, 8 dependent LDS loads each.
    if (tid < NB) {
      const int* rb = &buf[slot][0];
#pragma unroll
      for (int tl = 0; tl < TC; ++tl) {
        states[tl][tid] = s;                      // emitted state uses PRE-update s
        s = rb[(tl * NB + tid) * RS + s];         // s_{t+1} = bp[t, b, s_t]
      }
    }
    __syncthreads();

    // Gather tags at the now-known states: 128 independent scattered loads.
    if (tid < ROWS) {
      const int tl = tid / NB;
      const int bl = tid % NB;
      const int t  = c * TC + tl;
      const int b  = b0 + bl;
      const int sv = states[tl][bl];
      const float v =
          __builtin_nontemporal_load(tags + ((size_t)t * B + b) * NT + sv);
      out[(size_t)t * B + b] = v;
    }
    // No extra barrier needed: next chunk's staging targets the buffer the
    // chase already consumed, and states are rewritten only after the next
    // __syncthreads().
  }
}

extern "C" void kernel_launch(void* const* d_in, const int* in_sizes, int n_in,
                              void* d_out, int out_size, void* d_ws,
                              size_t ws_size, hipStream_t stream) {
  (void)in_sizes; (void)n_in; (void)out_size; (void)d_ws; (void)ws_size;
  const float* tags       = (const float*)d_in[0];  // (T, B, 128) f32
  const int*   bp         = (const int*)d_in[1];    // (T, B, 128) i32
  const int*   init_state = (const int*)d_in[2];    // (B,) i32
  float*       out        = (float*)d_out;          // (T, B, 1) f32

  dim3 grid(crf::B / crf::NB);  // 256 blocks x 16 chains
  dim3 block(256);
  crf_decode_kernel<<<grid, block, 0, stream>>>(tags, bp, init_state, out);
}